// SSIMLoss_3427383902593
// MI455X (gfx1250) — compile-verified
//
#include <hip/hip_runtime.h>

// ---------------------------------------------------------------------------
// SSIM loss for (128, 1, 512, 512) fp32 images, 11x11 Gaussian window.
// Conv passes run on the fp32 WMMA path (V_WMMA_F32_16X16X4_F32):
//   vertical:   V[16x32]   = G[16x28] * X[28x32]   (2 col groups)
//   horizontal: OutT[16x16]= G[16x28] * V_T[28x16] (V_T staged via LDS)
// G is the banded Gaussian matrix G[i][r] = w[r-i], r-i in [0,10]; it is the
// same matrix for both passes, so one set of A-operand constants serves all
// 105 WMMAs per tile. All 5 conv channels share the same C/D layout, so the
// elementwise SSIM math and the final mean are layout-independent.
// B-operand layout assumption (f32, 4x16): mirrors documented A layout —
//   VGPR0: lanes0-15 K=0, lanes16-31 K=2; VGPR1: K=1 / K=3; N = lane%16.
// ---------------------------------------------------------------------------

typedef float v2f __attribute__((ext_vector_type(2)));
typedef float v8f __attribute__((ext_vector_type(8)));

#define IMG_H 512
#define IMG_W 512
#define TS    16
#define KCH   7          // 7 K-chunks of 4 -> K = 28 (band 26 + 2 zero pad)
#define WAVES 4          // waves per block
#define NBLK  2048
#define NWAVE (NBLK * WAVES)

// Normalized 1-D Gaussian, sigma = 1.5, 11 taps.
__constant__ float WFULL[11] = {
    0.00102838f, 0.00759876f, 0.03600077f, 0.10936070f, 0.21300553f,
    0.26601172f,
    0.21300553f, 0.10936070f, 0.03600077f, 0.00759876f, 0.00102838f};

__device__ __forceinline__ float band_w(int i, int r) {
  int j = r - i;
  return (j >= 0 && j <= 10) ? WFULL[j] : 0.0f;
}

__device__ __forceinline__ float ldimg(const float* __restrict__ img, int iy,
                                       int ix) {
  return (iy >= 0 && iy < IMG_H && ix >= 0 && ix < IMG_W)
             ? img[iy * IMG_W + ix]
             : 0.0f;  // SAME padding == zero pad
}

__global__ void __launch_bounds__(WAVES * 32) ssim_tiles(
    const float* __restrict__ pred, const float* __restrict__ targ,
    float* __restrict__ partials, int totalTiles) {
  __shared__ float lds[WAVES][5][512];  // per-wave V_T staging: 5ch x 32 x 16
  __shared__ float wsum[WAVES];

  const int lane = threadIdx.x & 31;
  const int warp = threadIdx.x >> 5;
  const int m    = lane & 15;   // N (column) / M (row) index within half
  const int half = lane >> 4;   // lane half selects K pair / M+8
  float(*__restrict__ myLds)[512] = lds[warp];

  // A operand: banded Gaussian matrix chunks (constant, reused in both passes)
  // A layout (16x4 f32): lanes0-15 M=0..15 {K0,K1}; lanes16-31 M=0..15 {K2,K3}
  v2f aK[KCH];
#pragma unroll
  for (int k = 0; k < KCH; ++k) {
    const int base = 4 * k + 2 * half;
    aK[k].x = band_w(m, base);
    aK[k].y = band_w(m, base + 1);
  }

  const int wgid = blockIdx.x * WAVES + warp;
  float lsum = 0.0f;

  for (int t = wgid; t < totalTiles; t += NWAVE) {
    const int img = t >> 10;          // 1024 tiles per 512x512 image
    const int ty  = (t >> 5) & 31;
    const int tx  = t & 31;
    const float* pimg = pred + (size_t)img * (IMG_H * IMG_W);
    const float* timg = targ + (size_t)img * (IMG_H * IMG_W);

    // -------- Pass 1: vertical conv, V[16 x 32cols] = G * X --------
    v8f acc[5][2];
#pragma unroll
    for (int ch = 0; ch < 5; ++ch)
#pragma unroll
      for (int g = 0; g < 2; ++g) acc[ch][g] = v8f{};

#pragma unroll
    for (int k = 0; k < KCH; ++k) {
      const int r0 = 4 * k + 2 * half;  // halo-row pair this lane supplies
#pragma unroll
      for (int g = 0; g < 2; ++g) {
        const int ix  = tx * TS + g * 16 + m - 5;
        const int iy0 = ty * TS + r0 - 5;
        const float t0 = ldimg(timg, iy0, ix);
        const float t1 = ldimg(timg, iy0 + 1, ix);
        const float p0 = ldimg(pimg, iy0, ix);
        const float p1 = ldimg(pimg, iy0 + 1, ix);
        v2f b[5];
        b[0].x = t0;      b[0].y = t1;       // mu1 channel (target)
        b[1].x = p0;      b[1].y = p1;       // mu2 channel (prediction)
        b[2].x = t0 * t0; b[2].y = t1 * t1;  // E[t^2]
        b[3].x = p0 * p0; b[3].y = p1 * p1;  // E[p^2]
        b[4].x = t0 * p0; b[4].y = t1 * p1;  // E[t*p]
#pragma unroll
        for (int ch = 0; ch < 5; ++ch) {
          acc[ch][g] = __builtin_amdgcn_wmma_f32_16x16x4_f32(
              false, aK[k], false, b[ch], (short)0, acc[ch][g], false, false);
        }
      }
    }

    // -------- transpose V into LDS: V_T[c][i], c = g*16+N, i = M --------
    // C/D layout: lanes0-15 VGPR r -> (M=r, N=lane); lanes16-31 -> (M=r+8)
#pragma unroll
    for (int ch = 0; ch < 5; ++ch)
#pragma unroll
      for (int g = 0; g < 2; ++g)
#pragma unroll
        for (int r = 0; r < 8; ++r)
          myLds[ch][(g * 16 + m) * 16 + (r + 8 * half)] = acc[ch][g][r];

    // -------- Pass 2: horizontal conv, OutT[16x16] = G * V_T --------
    v8f out[5];
#pragma unroll
    for (int ch = 0; ch < 5; ++ch) out[ch] = v8f{};
#pragma unroll
    for (int k = 0; k < KCH; ++k) {
      const int c0 = 4 * k + 2 * half;
#pragma unroll
      for (int ch = 0; ch < 5; ++ch) {
        v2f b;
        b.x = myLds[ch][(c0 + 0) * 16 + m];
        b.y = myLds[ch][(c0 + 1) * 16 + m];
        out[ch] = __builtin_amdgcn_wmma_f32_16x16x4_f32(
            false, aK[k], false, b, (short)0, out[ch], false, false);
      }
    }

    // -------- elementwise SSIM map + local accumulation --------
    const float C1 = 1e-4f, C2 = 9e-4f;
#pragma unroll
    for (int r = 0; r < 8; ++r) {
      const float mu1 = out[0][r];
      const float mu2 = out[1][r];
      const float m11 = mu1 * mu1, m22 = mu2 * mu2, m12 = mu1 * mu2;
      const float s11 = out[2][r] - m11;   // sigma1^2
      const float s22 = out[3][r] - m22;   // sigma2^2
      const float s12 = out[4][r] - m12;   // sigma12
      const float num = (2.0f * m12 + C1) * (2.0f * s12 + C2);
      const float den = (m11 + m22 + C1) * (s11 + s22 + C2);
      lsum += num / den;
    }
  }

  // deterministic reduction: wave shuffle -> block partial -> d_ws slot
#pragma unroll
  for (int off = 16; off > 0; off >>= 1) lsum += __shfl_xor(lsum, off, 32);
  if (lane == 0) wsum[warp] = lsum;
  __syncthreads();
  if (threadIdx.x == 0) {
    float bs = 0.0f;
#pragma unroll
    for (int wv = 0; wv < WAVES; ++wv) bs += wsum[wv];
    partials[blockIdx.x] = bs;
  }
}

__global__ void ssim_finalize(const float* __restrict__ partials,
                              float* __restrict__ outp, int nPart,
                              double invN) {
  if (threadIdx.x == 0 && blockIdx.x == 0) {
    double s = 0.0;
    for (int i = 0; i < nPart; ++i) s += (double)partials[i];
    outp[0] = (float)(1.0 - s * invN);
  }
}

extern "C" void kernel_launch(void* const* d_in, const int* in_sizes, int n_in,
                              void* d_out, int out_size, void* d_ws,
                              size_t ws_size, hipStream_t stream) {
  const float* pred = (const float*)d_in[0];  // prediction
  const float* targ = (const float*)d_in[1];  // target
  float* partials = (float*)d_ws;             // NBLK floats = 8 KB
  float* outp = (float*)d_out;

  const int nImg = in_sizes[0] / (IMG_H * IMG_W);           // 128
  const int totalTiles = nImg * (IMG_H / TS) * (IMG_W / TS);  // 131072

  ssim_tiles<<<NBLK, WAVES * 32, 0, stream>>>(pred, targ, partials, totalTiles);

  const double invN = 1.0 / ((double)nImg * IMG_H * IMG_W);
  ssim_finalize<<<1, 32, 0, stream>>>(partials, outp, NBLK, invN);
}